// GCNLayer_87522843557970
// MI455X (gfx1250) — compile-verified
//
#include <hip/hip_runtime.h>
#include <hip/hip_bf16.h>
#include <stdint.h>

// Problem constants (match the reference)
#define B_    16
#define NG_   4
#define G_    256
#define DIN_  834
#define S_    1568   // 8*14*14
#define IN_   1024
#define NMAX_ 16

typedef float v2f __attribute__((ext_vector_type(2)));
typedef float v8f __attribute__((ext_vector_type(8)));

// CDNA5 fp32 tensor op: D(16x16,f32) = A(16x4,f32) * B(4x16,f32) + C
__device__ __forceinline__ v8f wmma_f32_k4(v2f a, v2f b, v8f c) {
    return __builtin_amdgcn_wmma_f32_16x16x4_f32(
        /*neg_a=*/false, a, /*neg_b=*/false, b,
        /*c_mod=*/(short)0, c, /*reuse_a=*/false, /*reuse_b=*/false);
}

// Lane decomposition for 16x16x4 f32 WMMA (wave32):
//   lane = half*16 + lm ;  A frag: lane holds (M=lm, K=klo..klo+1), klo = half*2
//   B frag: lane holds (N=lm, K=klo..klo+1)
//   C/D  : VGPR r -> M = half*8 + r, N = lm

// ---------------------------------------------------------------------------
// K1: emb = actor_features @ W_a^T + b_a, padded/zero-masked into [B,NG,16,G]
// ---------------------------------------------------------------------------
__global__ __launch_bounds__(256)
void k1_embed_pad(const float* __restrict__ A,        // [total, IN]
                  const float* __restrict__ Wa,       // [NG, G, IN]
                  const float* __restrict__ ba,       // [NG, G]
                  const long long* __restrict__ counts,
                  float* __restrict__ padded)         // [B, NG, NMAX, G]
{
    int tile = blockIdx.x * (blockDim.x >> 5) + (threadIdx.x >> 5);
    int ft = tile & 15;
    int g  = (tile >> 4) & 3;
    int b  = tile >> 6;
    int lane = threadIdx.x & 31;
    int half = lane >> 4, lm = lane & 15, klo = half * 2;

    long long off = 0;
    for (int i = 0; i < b; ++i) off += counts[i];
    int cnt = (int)counts[b];

    int m = lm;
    long long arow = off + (m < cnt ? m : cnt - 1);
    const float* Arow = A + arow * (long long)IN_;
    int fbase = ft * 16;
    const float* Brow = Wa + ((long long)g * G_ + (fbase + lm)) * IN_;

    v8f c = {0.f,0.f,0.f,0.f,0.f,0.f,0.f,0.f};
    for (int k0 = 0; k0 < IN_; k0 += 4) {
        v2f a;  a.x  = Arow[k0 + klo]; a.y  = Arow[k0 + klo + 1];
        v2f bb; bb.x = Brow[k0 + klo]; bb.y = Brow[k0 + klo + 1];
        c = wmma_f32_k4(a, bb, c);
    }

    float bias = ba[g * G_ + fbase + lm];
    float* outb = padded + ((long long)b * NG_ + g) * (NMAX_ * G_);
#pragma unroll
    for (int r = 0; r < 8; ++r) {
        int mr = half * 8 + r;
        float v = (mr < cnt) ? (c[r] + bias) : 0.f;
        outb[mr * G_ + fbase + lm] = v;
    }
}

// ---------------------------------------------------------------------------
// K2: ctx[b,g,f,s] = sum_c W_c[g,f,c] * fm[b,c,s] + b_c[g,f]
// Dominant GEMM. fm k-slabs are shared by all 8 waves of a workgroup, so they
// are staged once into LDS with double-buffered GLOBAL_LOAD_ASYNC_TO_LDS_B128
// (ASYNCcnt), overlapping the copy of chunk k+1 with WMMA on chunk k.
// Chunk = 8 c-rows x 128 floats (row-padded from 112) = 4 KB = 256 thr x 16 B.
// ---------------------------------------------------------------------------
__global__ __launch_bounds__(256)
void k2_ctx(const float* __restrict__ fm,   // [B, DIN, S]
            const float* __restrict__ Wc,   // [NG, G, DIN]
            const float* __restrict__ bc,   // [NG, G]
            float* __restrict__ ctx)        // [B, NG, G, S]
{
    __shared__ __attribute__((aligned(16))) float fmS[2][8][128];

    int bg = blockIdx.z;
    int g = bg & 3, b = bg >> 2;
    int tid  = threadIdx.x;
    int wave = tid >> 5;
    int lane = tid & 31;
    int half = lane >> 4, lm = lane & 15, klo = half * 2;

    int fbase = blockIdx.y * 128 + wave * 16;
    int sbase = blockIdx.x * 112;

    const float* fmb  = fm + (long long)b * DIN_ * S_;
    const float* Arow = Wc + ((long long)g * G_ + fbase + lm) * DIN_;

    // one 16B async packet per thread per chunk (clamped in s and c)
    int crow = tid >> 5;                 // 0..7  c-row within chunk
    int ccol = lane * 4;                 // 0..124 float column within slab
    int scol = sbase + ccol;
    if (scol > S_ - 4) scol = S_ - 4;    // stay in-bounds (extras never read)

    v8f acc[7];
#pragma unroll
    for (int j = 0; j < 7; ++j) {
        v8f z = {0.f,0.f,0.f,0.f,0.f,0.f,0.f,0.f};
        acc[j] = z;
    }

    const int NCH = (DIN_ + 7) / 8;      // 105 chunks (last has 2 valid rows)

    auto issue_chunk = [&](int kc, int buf) {
        int c = kc * 8 + crow;
        if (c > DIN_ - 1) c = DIN_ - 1;  // clamp; tail garbage is A-masked
        const float* src = fmb + (long long)c * S_ + scol;
        unsigned lds_off =
            (unsigned)(uintptr_t)&fmS[buf][crow][ccol];   // low 32b = LDS addr
        asm volatile("global_load_async_to_lds_b128 %0, %1, off"
                     :: "v"(lds_off), "v"((unsigned long long)(uintptr_t)src)
                     : "memory");
    };

    issue_chunk(0, 0);
    for (int kc = 0; kc < NCH; ++kc) {
        int cur = kc & 1;
        if (kc + 1 < NCH) {
            issue_chunk(kc + 1, cur ^ 1);
            asm volatile("s_wait_asynccnt 0x1" ::: "memory"); // chunk kc landed
        } else {
            asm volatile("s_wait_asynccnt 0x0" ::: "memory");
        }
        __syncthreads();                 // slab visible to all 8 waves

        int c0 = kc * 8;
        __builtin_prefetch(Arow + c0 + 32, 0, 3);            // global_prefetch
        if (kc < NCH - 1) {
#pragma unroll
            for (int ks = 0; ks < 8; ks += 4) {
                v2f a; a.x = Arow[c0 + ks + klo];
                       a.y = Arow[c0 + ks + klo + 1];
#pragma unroll
                for (int j = 0; j < 7; ++j) {
                    v2f bb; bb.x = fmS[cur][ks + klo][j * 16 + lm];
                            bb.y = fmS[cur][ks + klo + 1][j * 16 + lm];
                    acc[j] = wmma_f32_k4(a, bb, acc[j]);
                }
            }
        } else {
            // K tail (834 = 104*8 + 2): zero A where K is OOB
#pragma unroll
            for (int ks = 0; ks < 8; ks += 4) {
                v2f a;
                a.x = (c0 + ks + klo     < DIN_) ? Arow[c0 + ks + klo]     : 0.f;
                a.y = (c0 + ks + klo + 1 < DIN_) ? Arow[c0 + ks + klo + 1] : 0.f;
#pragma unroll
                for (int j = 0; j < 7; ++j) {
                    v2f bb; bb.x = fmS[cur][ks + klo][j * 16 + lm];
                            bb.y = fmS[cur][ks + klo + 1][j * 16 + lm];
                    acc[j] = wmma_f32_k4(a, bb, acc[j]);
                }
            }
        }
        __syncthreads();                 // before next chunk overwrites buffer
    }

    float* outb = ctx + ((long long)b * NG_ + g) * (long long)(G_ * S_);
#pragma unroll
    for (int r = 0; r < 8; ++r) {
        int f = fbase + half * 8 + r;
        float bias = bc[g * G_ + f];
#pragma unroll
        for (int j = 0; j < 7; ++j)
            outb[(long long)f * S_ + sbase + j * 16 + lm] = acc[j][r] + bias;
    }
}

// ---------------------------------------------------------------------------
// K3: fused per (b,g):  logits = padded @ ctx  -> masked softmax over S
//                       upd    = adj @ ctx^T + padded
// ---------------------------------------------------------------------------
__global__ __launch_bounds__(256)
void k3_adj_softmax_upd(const float* __restrict__ ctx,     // [B,NG,G,S]
                        const float* __restrict__ padded,  // [B,NG,NMAX,G]
                        const long long* __restrict__ counts,
                        float* __restrict__ upd)           // [B,NG,NMAX,G]
{
    __shared__ float Apad[NMAX_][G_];       //  16 KB
    __shared__ float Ladj[NMAX_][S_ + 8];   // ~101 KB

    int bg = blockIdx.x;
    int g = bg & 3, b = bg >> 2;
    int tid = threadIdx.x, wave = tid >> 5, lane = tid & 31;
    int half = lane >> 4, lm = lane & 15, klo = half * 2;
    int cnt = (int)counts[b];

    const float* padb = padded + ((long long)b * NG_ + g) * (NMAX_ * G_);
    const float* ctxb = ctx + ((long long)b * NG_ + g) * (long long)(G_ * S_);

    for (int i = tid; i < NMAX_ * G_; i += 256)
        Apad[i >> 8][i & 255] = padb[i];
    __syncthreads();

    // ---- logits[n, s] = sum_f Apad[n,f] * ctx[f,s]
    for (int st = wave; st < S_ / 16; st += 8) {
        int sb = st * 16;
        v8f c = {0.f,0.f,0.f,0.f,0.f,0.f,0.f,0.f};
        for (int f0 = 0; f0 < G_; f0 += 4) {
            v2f a;  a.x  = Apad[lm][f0 + klo];
                    a.y  = Apad[lm][f0 + klo + 1];
            v2f bb; bb.x = ctxb[(long long)(f0 + klo)     * S_ + sb + lm];
                    bb.y = ctxb[(long long)(f0 + klo + 1) * S_ + sb + lm];
            c = wmma_f32_k4(a, bb, c);
        }
#pragma unroll
        for (int r = 0; r < 8; ++r)
            Ladj[half * 8 + r][sb + lm] = c[r];
    }
    __syncthreads();

    // ---- masked softmax over S (wave32 shuffle reductions)
    for (int n = wave; n < NMAX_; n += 8) {
        float mx = -3.402823466e38f;
        for (int s = lane; s < S_; s += 32) mx = fmaxf(mx, Ladj[n][s]);
        for (int o = 16; o; o >>= 1) mx = fmaxf(mx, __shfl_xor(mx, o, 32));
        float sum = 0.f;
        for (int s = lane; s < S_; s += 32) {
            float e = __expf(Ladj[n][s] - mx);
            Ladj[n][s] = e;
            sum += e;
        }
        for (int o = 16; o; o >>= 1) sum += __shfl_xor(sum, o, 32);
        float scale = (n < cnt) ? (1.f / sum) : 0.f;
        for (int s = lane; s < S_; s += 32) Ladj[n][s] *= scale;
    }
    __syncthreads();

    // ---- upd[n, f] = sum_s adj[n,s] * ctx[f,s] + padded[n,f]
    float* updb = upd + ((long long)b * NG_ + g) * (NMAX_ * G_);
    for (int ftile = wave; ftile < 16; ftile += 8) {
        int fb = ftile * 16;
        v8f c = {0.f,0.f,0.f,0.f,0.f,0.f,0.f,0.f};
        const float* crow = ctxb + (long long)(fb + lm) * S_;
        for (int s0 = 0; s0 < S_; s0 += 4) {
            v2f a;  a.x  = Ladj[lm][s0 + klo];
                    a.y  = Ladj[lm][s0 + klo + 1];
            v2f bb; bb.x = crow[s0 + klo];
                    bb.y = crow[s0 + klo + 1];
            c = wmma_f32_k4(a, bb, c);
        }
#pragma unroll
        for (int r = 0; r < 8; ++r) {
            int n = half * 8 + r;
            updb[n * G_ + fb + lm] = c[r] + Apad[n][fb + lm];
        }
    }
}

// ---------------------------------------------------------------------------
// K4: out[t, g*G+h] = relu( sum_f upd[b,g,n,f] * W_h[g,h,f] ), ragged scatter
// ---------------------------------------------------------------------------
__global__ __launch_bounds__(256)
void k4_head(const float* __restrict__ upd,   // [B,NG,NMAX,G]
             const float* __restrict__ Wh,    // [NG, G(h), G(f)]
             const long long* __restrict__ counts,
             float* __restrict__ out)         // [total, NG*G]
{
    int tile = blockIdx.x * (blockDim.x >> 5) + (threadIdx.x >> 5);
    int ht = tile & 15;
    int g  = (tile >> 4) & 3;
    int b  = tile >> 6;
    int lane = threadIdx.x & 31;
    int half = lane >> 4, lm = lane & 15, klo = half * 2;

    long long off = 0;
    for (int i = 0; i < b; ++i) off += counts[i];
    int cnt = (int)counts[b];

    const float* Ab   = upd + ((long long)b * NG_ + g) * (NMAX_ * G_);
    const float* Brow = Wh + ((long long)g * G_ + ht * 16 + lm) * G_;

    v8f c = {0.f,0.f,0.f,0.f,0.f,0.f,0.f,0.f};
    for (int k0 = 0; k0 < G_; k0 += 4) {
        v2f a;  a.x  = Ab[lm * G_ + k0 + klo];
                a.y  = Ab[lm * G_ + k0 + klo + 1];
        v2f bb; bb.x = Brow[k0 + klo];
                bb.y = Brow[k0 + klo + 1];
        c = wmma_f32_k4(a, bb, c);
    }
#pragma unroll
    for (int r = 0; r < 8; ++r) {
        int n = half * 8 + r;
        if (n < cnt) {
            long long t = off + n;
            out[t * (NG_ * G_) + g * G_ + ht * 16 + lm] = fmaxf(c[r], 0.f);
        }
    }
}

// ---------------------------------------------------------------------------
extern "C" void kernel_launch(void* const* d_in, const int* in_sizes, int n_in,
                              void* d_out, int out_size, void* d_ws, size_t ws_size,
                              hipStream_t stream) {
    const float*     actor  = (const float*)d_in[0];   // [total, 1024]
    const float*     fm     = (const float*)d_in[1];   // [16, 834, 1568]
    const float*     Wa     = (const float*)d_in[2];   // [4, 256, 1024]
    const float*     ba     = (const float*)d_in[3];   // [4, 256]
    const float*     Wc     = (const float*)d_in[4];   // [4, 256, 834]
    const float*     bc     = (const float*)d_in[5];   // [4, 256]
    const float*     Wh     = (const float*)d_in[6];   // [4, 256, 256]
    const long long* counts = (const long long*)d_in[7]; // [16] int64
    float* out = (float*)d_out;

    float* ws     = (float*)d_ws;
    float* padded = ws;                                        // 262,144 f32
    float* ctx    = padded + (size_t)B_ * NG_ * NMAX_ * G_;    // 25,690,112 f32
    float* updb   = ctx + (size_t)B_ * NG_ * G_ * S_;          // 262,144 f32

    // K1: 1024 wave-tiles, 8 waves/block
    k1_embed_pad<<<128, 256, 0, stream>>>(actor, Wa, ba, counts, padded);

    // K2: grid (s-blocks=14, f-blocks=2, b*g=64)
    dim3 g2(14, 2, 64);
    k2_ctx<<<g2, 256, 0, stream>>>(fm, Wc, bc, ctx);

    // K3: one workgroup per (b,g)
    k3_adj_softmax_upd<<<64, 256, 0, stream>>>(ctx, padded, counts, updb);

    // K4: 1024 wave-tiles, 8 waves/block
    k4_head<<<128, 256, 0, stream>>>(updb, Wh, counts, out);
}